// UFGConv_90744069030461
// MI455X (gfx1250) — compile-verified
//
#include <hip/hip_runtime.h>
#include <math.h>

#define F_DIM 64

typedef __attribute__((ext_vector_type(2))) float v2f;
typedef __attribute__((ext_vector_type(8))) float v8f;

// ---------------- GEMM: h = x @ W via V_WMMA_F32_16X16X4_F32 ----------------
// grid.x = N/16 row tiles, block = 128 (4 waves -> 4 column tiles of 16)
__global__ void __launch_bounds__(128)
gemm_wmma_f32(float* __restrict__ h, const float* __restrict__ x,
              const float* __restrict__ w) {
  const int lane = threadIdx.x & 31;
  const int wv   = threadIdx.x >> 5;     // wave id 0..3 -> 16-col tile
  const int half = lane >> 4;            // 0: K pair {k0,k0+1}; 1: {k0+2,k0+3}
  const int lm   = lane & 15;
  const long long rowBase = (long long)blockIdx.x * 16;
  const int colBase = wv * 16;
  const float* xr = x + (rowBase + lm) * F_DIM;
  v8f acc = {};
#pragma unroll
  for (int k0 = 0; k0 < F_DIM; k0 += 4) {
    const int ka = k0 + 2 * half;
    v2f a, b;
    // A 16x4 layout: VGPR0 = K even-of-pair, VGPR1 = next K; lanes 0-15 M rows
    a.x = xr[ka];
    a.y = xr[ka + 1];
    // B 4x16 layout (mirror): row K striped across lanes within a VGPR
    b.x = w[ka * F_DIM + colBase + lm];
    b.y = w[(ka + 1) * F_DIM + colBase + lm];
    acc = __builtin_amdgcn_wmma_f32_16x16x4_f32(false, a, false, b,
                                                (short)0, acc, false, false);
  }
#pragma unroll
  for (int v = 0; v < 8; ++v) {
    const int m = v + 8 * half;          // C/D: lanes>=16 hold rows 8..15
    h[(rowBase + m) * F_DIM + colBase + lm] = acc[v];
  }
}

// ---------------- derived scalars: p[l] = s^(-J+l) * (2/pi) ----------------
__global__ void setup_scales(float* __restrict__ p, const float* __restrict__ s,
                             const float* __restrict__ J) {
  if (threadIdx.x == 0 && blockIdx.x == 0) {
    const float sv = s[0], Jv = J[0];
    const float invA = 2.0f / 3.14159265358979323846f;   // 1/(pi/2)
    p[0] = powf(sv, -Jv) * invA;          // level 1: scale/A
    p[1] = powf(sv, -Jv + 1.0f) * invA;   // level 2: scale/A
  }
}

// ---------------- elementwise helpers ----------------
__global__ void scale_copy(float* __restrict__ dst, const float* __restrict__ src,
                           float a, long long n) {
  long long i = (long long)blockIdx.x * blockDim.x + threadIdx.x;
  if (i < n) dst[i] = a * src[i];
}

__global__ void axpby_write(float* __restrict__ dst, const float* __restrict__ a,
                            const float* __restrict__ b, float ca, float cb,
                            long long n) {
  long long i = (long long)blockIdx.x * blockDim.x + threadIdx.x;
  if (i < n) dst[i] = ca * a[i] + cb * b[i];
}

// SpMM scatter: y[row,f] += alpha * val[e] * x[col,f]; one lane per (edge,f)
__global__ void spmm_scatter(float* __restrict__ y, const float* __restrict__ x,
                             const int* __restrict__ rows,
                             const int* __restrict__ cols,
                             const float* __restrict__ vals,
                             const float* __restrict__ alphaP, float factor,
                             long long total) {
  long long g = (long long)blockIdx.x * blockDim.x + threadIdx.x;
  if (g >= total) return;
  const int e = (int)(g >> 6);
  const int f = (int)(g & (F_DIM - 1));
  const float alpha = factor * alphaP[0];
  const int r = rows[e];
  const int cidx = cols[e];
  atomicAdd(&y[(long long)r * F_DIM + f],
            alpha * vals[e] * x[(long long)cidx * F_DIM + f]);
}

// D_j = 0.5*c[j,0]*T0 + c[j,1]*T1 + c[j,2]*T2  (both filters in one pass)
__global__ void decomp_combine(float* __restrict__ D0, float* __restrict__ D1,
                               const float* __restrict__ T0,
                               const float* __restrict__ T1,
                               const float* __restrict__ T2,
                               const float* __restrict__ c, long long n) {
  long long i = (long long)blockIdx.x * blockDim.x + threadIdx.x;
  if (i >= n) return;
  const float t0 = T0[i], t1 = T1[i], t2 = T2[i];
  D0[i] = 0.5f * c[0] * t0 + c[1] * t1 + c[2] * t2;
  D1[i] = 0.5f * c[3] * t0 + c[4] * t1 + c[5] * t2;
}

// soft shrink then per-node filter scale: D = sign(D)*relu(|D|-t) * filt[node]
__global__ void shrink_scale(float* __restrict__ D, const float* __restrict__ filt,
                             long long n) {
  long long i = (long long)blockIdx.x * blockDim.x + threadIdx.x;
  if (i >= n) return;
  const float v = D[i];
  const float a = fmaxf(fabsf(v) - 1e-4f, 0.0f);
  D[i] = copysignf(a, v) * filt[i >> 6];
}

// acc += 0.5*cj[0]*T0 + cj[1]*T1 + cj[2]*T2
__global__ void recon_accum(float* __restrict__ acc, const float* __restrict__ T0,
                            const float* __restrict__ T1,
                            const float* __restrict__ T2,
                            const float* __restrict__ cj, long long n) {
  long long i = (long long)blockIdx.x * blockDim.x + threadIdx.x;
  if (i >= n) return;
  acc[i] += 0.5f * cj[0] * T0[i] + cj[1] * T1[i] + cj[2] * T2[i];
}

__global__ void bias_add(float* __restrict__ out, const float* __restrict__ src,
                         const float* __restrict__ bias, long long n) {
  long long i = (long long)blockIdx.x * blockDim.x + threadIdx.x;
  if (i < n) out[i] = src[i] + bias[i & (F_DIM - 1)];
}

extern "C" void kernel_launch(void* const* d_in, const int* in_sizes, int n_in,
                              void* d_out, int out_size, void* d_ws, size_t ws_size,
                              hipStream_t stream) {
  const float* x      = (const float*)d_in[0];
  const int*   Lrow   = (const int*)d_in[1];          // L_index[0] = rows
  const float* Lval   = (const float*)d_in[2];
  const float* c      = (const float*)d_in[3];        // [2,3]
  const float* sP     = (const float*)d_in[4];
  const float* JP     = (const float*)d_in[5];
  const float* weight = (const float*)d_in[6];        // [64,64]
  const float* filt   = (const float*)d_in[7];        // [4*N,1]
  const float* bias   = (const float*)d_in[8];

  const int E = in_sizes[2];
  const int N = in_sizes[0] / F_DIM;
  const int* Lcol = Lrow + E;                         // L_index[1] = cols
  const long long NB = (long long)N * F_DIM;

  float* params = (float*)d_ws;                       // 64-float scalar slot
  float* base = params + 64;
  float* B0 = base;            // h; later recon level-2 accumulator (FD_rec)
  float* B1 = B0 + NB;         // T1
  float* B2 = B1 + NB;         // T2
  float* B3 = B2 + NB;         // D[0]
  float* B4 = B3 + NB;         // D[1]
  float* B5 = B4 + NB;         // D[2]; later recon level-1 accumulator
  float* B6 = B5 + NB;         // D[3]

  const int BLK = 256;
  const int ewG = (int)((NB + BLK - 1) / BLK);
  const long long spTot = (long long)E * F_DIM;
  const int spG = (int)((spTot + BLK - 1) / BLK);
  const float* a1 = params + 0;   // level-1 scale/A
  const float* a2 = params + 1;   // level-2 scale/A

  setup_scales<<<1, 32, 0, stream>>>(params, sP, JP);
  gemm_wmma_f32<<<dim3(N / 16), dim3(128), 0, stream>>>(B0, x, weight);

  // One Chebyshev step: T1 = a*L*T0 - T0 ; T2 = 2a*L*T1 - 2*T1 - T0
#define CHEB(T0P, AP)                                                          \
  scale_copy<<<ewG, BLK, 0, stream>>>(B1, (T0P), -1.0f, NB);                   \
  spmm_scatter<<<spG, BLK, 0, stream>>>(B1, (T0P), Lrow, Lcol, Lval, (AP),     \
                                        1.0f, spTot);                          \
  axpby_write<<<ewG, BLK, 0, stream>>>(B2, B1, (T0P), -2.0f, -1.0f, NB);       \
  spmm_scatter<<<spG, BLK, 0, stream>>>(B2, B1, Lrow, Lcol, Lval, (AP),        \
                                        2.0f, spTot)

  // ---- decomp level 1 (T0 = h) ----
  CHEB(B0, a1);
  decomp_combine<<<ewG, BLK, 0, stream>>>(B3, B4, B0, B1, B2, c, NB);
  // ---- decomp level 2 (T0 = unshrunk D[0]) ----
  CHEB(B3, a2);
  decomp_combine<<<ewG, BLK, 0, stream>>>(B5, B6, B3, B1, B2, c, NB);

  // ---- shrink + filt per block ----
  shrink_scale<<<ewG, BLK, 0, stream>>>(B3, filt + 0LL * N, NB);
  shrink_scale<<<ewG, BLK, 0, stream>>>(B4, filt + 1LL * N, NB);
  shrink_scale<<<ewG, BLK, 0, stream>>>(B5, filt + 2LL * N, NB);
  shrink_scale<<<ewG, BLK, 0, stream>>>(B6, filt + 3LL * N, NB);

  // ---- recon level 2: acc = B0 (h dead), T0 = D[2] then D[3] ----
  hipMemsetAsync(B0, 0, NB * sizeof(float), stream);
  CHEB(B5, a2);
  recon_accum<<<ewG, BLK, 0, stream>>>(B0, B5, B1, B2, c + 0, NB);
  CHEB(B6, a2);
  recon_accum<<<ewG, BLK, 0, stream>>>(B0, B6, B1, B2, c + 3, NB);

  // ---- recon level 1: acc = B5 (D[2] dead), T0 = FD_rec then D[1] ----
  hipMemsetAsync(B5, 0, NB * sizeof(float), stream);
  CHEB(B0, a1);
  recon_accum<<<ewG, BLK, 0, stream>>>(B5, B0, B1, B2, c + 0, NB);
  CHEB(B4, a1);
  recon_accum<<<ewG, BLK, 0, stream>>>(B5, B4, B1, B2, c + 3, NB);

  bias_add<<<ewG, BLK, 0, stream>>>((float*)d_out, B5, bias, NB);
#undef CHEB
}